// UnlikelihoodLoss_31817117729134
// MI455X (gfx1250) — compile-verified
//
#include <hip/hip_runtime.h>

// ---------------- problem constants (match reference) ----------------
#define BB 2
#define SS 1024
#define VV 32000
#define ROWS (BB * SS)          // 2048 token rows
#define TILE 4000               // floats per LDS tile (16 KB)
#define NTILE 8                 // 8 * 4000 = 32000 = VV
#define CHUNKS 1000             // float4 chunks per tile
#define THREADS 256             // 8 waves of 32

// =====================================================================
// Kernel 1: one block per token row.
//  - stream the 32000-float row through LDS with double-buffered
//    GLOBAL_LOAD_ASYNC_TO_LDS_B128 (ASYNCcnt-tracked CDNA5 path)
//  - exact LSE via per-tile (max, expsum) merge; plain sum for smoothing
//  - unlikelihood candidates from a 127-label window, deduped in LDS
//  - per-row partials written to workspace (deterministic order)
// =====================================================================
__global__ __launch_bounds__(THREADS) void ul_row_kernel(
    const float* __restrict__ logits,
    const int* __restrict__ labels,
    float* __restrict__ ws) {
  __shared__ __align__(16) float smem[2 * TILE];  // two 16KB tile buffers
  __shared__ float red[16];                       // cross-wave scratch
  __shared__ int wl[127];                         // window labels

  const int row = blockIdx.x;
  const int b = row / SS;
  const int i = row - b * SS;
  const float* rowp = logits + (size_t)row * VV;
  const int tid = threadIdx.x;
  const int lane = tid & 31;
  const int wid = tid >> 5;

  // Issue one tile's async DMA: every thread issues exactly 4 loads so the
  // per-wave ASYNCcnt increment is a known constant (4). Tail lanes clamp
  // to a duplicate chunk (same data to same LDS bytes -> benign).
  auto issue_tile = [&](int t) {
    const float* gbase = rowp + t * TILE;
    float* lbase = smem + (t & 1) * TILE;
#pragma unroll
    for (int k = 0; k < 4; ++k) {
      int c = tid + THREADS * k;
      c = (c < CHUNKS) ? c : (CHUNKS - 1);
      // low 32 bits of a flat pointer into LDS == LDS byte offset (ISA 10.2)
      unsigned loff = (unsigned)(size_t)(lbase + c * 4);
      const float* g = gbase + c * 4;
      asm volatile("global_load_async_to_lds_b128 %0, %1, off"
                   :: "v"(loff), "v"(g)
                   : "memory");
    }
  };

  float M = -__builtin_inff();  // running row max
  float S = 0.0f;               // running sum exp(x - M)
  float P = 0.0f;               // running plain sum (for smoothing mean)

  issue_tile(0);
  for (int t = 0; t < NTILE; ++t) {
    if (t + 1 < NTILE) {
      issue_tile(t + 1);
      // tile t+1 in flight (4 outstanding); wait until tile t retired
      asm volatile("s_wait_asynccnt 0x4" ::: "memory");
    } else {
      asm volatile("s_wait_asynccnt 0x0" ::: "memory");
    }
    __syncthreads();  // all waves' DMA for tile t visible in LDS

    const float4* buf = (const float4*)(smem + (t & 1) * TILE);

    // ---- phase A: tile max + tile plain sum (from LDS) ----
    float vmax = -__builtin_inff(), vsum = 0.0f;
    for (int c = tid; c < CHUNKS; c += THREADS) {
      float4 x = buf[c];
      vmax = fmaxf(fmaxf(vmax, fmaxf(x.x, x.y)), fmaxf(x.z, x.w));
      vsum += (x.x + x.y) + (x.z + x.w);
    }
    for (int o = 16; o > 0; o >>= 1) {
      vmax = fmaxf(vmax, __shfl_down(vmax, o, 32));
      vsum += __shfl_down(vsum, o, 32);
    }
    if (lane == 0) { red[wid] = vmax; red[8 + wid] = vsum; }
    __syncthreads();
    float m_t = red[0], p_t = red[8];
    for (int w = 1; w < 8; ++w) {
      m_t = fmaxf(m_t, red[w]);
      p_t += red[8 + w];
    }
    __syncthreads();

    // ---- phase B: tile exp-sum vs tile max (one exp per element) ----
    float esum = 0.0f;
    for (int c = tid; c < CHUNKS; c += THREADS) {
      float4 x = buf[c];
      esum += __expf(x.x - m_t) + __expf(x.y - m_t) +
              __expf(x.z - m_t) + __expf(x.w - m_t);
    }
    for (int o = 16; o > 0; o >>= 1) esum += __shfl_down(esum, o, 32);
    if (lane == 0) red[wid] = esum;
    __syncthreads();
    float s_t = red[0];
    for (int w = 1; w < 8; ++w) s_t += red[w];
    __syncthreads();  // scratch free before next tile / buffer reuse

    // ---- merge tile stats into running LSE state ----
    float nM = fmaxf(M, m_t);
    S = S * __expf(M - nM) + s_t * __expf(m_t - nM);
    M = nM;
    P += p_t;
  }
  const float lse = M + __logf(S);

  // ---- unlikelihood: unique previous-window labels, != target, != 0 ----
  const int* labrow = labels + b * SS;
  const int lab = labrow[i];
  if (tid < 127) {
    int j = i - 1 - tid;                 // k = tid+1 in (i-128, i)
    int c = (j >= 0) ? labrow[j] : 0;
    if (c == lab) c = 0;                 // drop candidates equal to target
    wl[tid] = c;
  }
  __syncthreads();
  float uo = 0.0f;
  if (tid < 127) {
    int c = wl[tid];
    bool keep = (c != 0);
    if (keep) {
      for (int t2 = 0; t2 < tid; ++t2) {
        if (wl[t2] == c) { keep = false; break; }  // dedupe (nearest wins)
      }
    }
    if (keep) {
      float p = __expf(rowp[c] - lse);              // softmax[c], L2-hot gather
      uo = -__logf(fmaxf(1.0f - p, 1e-5f));
    }
  }
  for (int o = 16; o > 0; o >>= 1) uo += __shfl_down(uo, o, 32);
  if (lane == 0) red[wid] = uo;
  __syncthreads();

  if (tid == 0) {
    float u_tok = 0.0f;
    for (int w = 0; w < 8; ++w) u_tok += red[w];
    float xt = rowp[lab];
    float nll = lse - xt;
    float smooth = lse - P * (1.0f / (float)VV);
    float per_tok = 0.9f * nll + 0.1f * smooth;     // label_smoothing = 0.1
    float valid = (lab != 0) ? 1.0f : 0.0f;         // ignore_index = 0
    ws[row] = per_tok * valid;
    ws[ROWS + row] = valid;
    ws[2 * ROWS + row] = u_tok;
  }
}

// =====================================================================
// Kernel 2: single-wave deterministic reduction of the 3x2048 partials
// using V_WMMA_F32_16X16X4_F32 with a ones B-matrix:
//   D[m,n] += sum_k A[m,k] * 1   -> every column holds row sums.
// 2048 = 32 iterations of 16x4 blocks per array.
// =====================================================================
typedef float v2f __attribute__((ext_vector_type(2)));
typedef float v8f __attribute__((ext_vector_type(8)));

__global__ __launch_bounds__(32) void ul_finalize_kernel(
    const float* __restrict__ ws, float* __restrict__ out) {
  const int lane = threadIdx.x;        // one wave32, EXEC all ones
  const int mrow = lane & 15;          // A row M
  const int kbase = (lane >> 4) * 2;   // lanes 0-15: K=0,1  lanes 16-31: K=2,3
  v2f ones = {1.0f, 1.0f};

  float tot[3];
#pragma unroll
  for (int a = 0; a < 3; ++a) {
    const float* p = ws + a * ROWS;
    v8f acc = {0.f, 0.f, 0.f, 0.f, 0.f, 0.f, 0.f, 0.f};
    for (int t = 0; t < 32; ++t) {
      int base = t * 64 + mrow * 4 + kbase;   // A[m,k] = p[t*64 + m*4 + k]
      v2f av;
      av.x = p[base];
      av.y = p[base + 1];
      acc = __builtin_amdgcn_wmma_f32_16x16x4_f32(
          false, av, false, ones, (short)0, acc, false, false);
    }
    // column 0: lane 0 holds M=0..7, lane 16 holds M=8..15 (one per VGPR)
    float s = ((acc[0] + acc[1]) + (acc[2] + acc[3])) +
              ((acc[4] + acc[5]) + (acc[6] + acc[7]));
    tot[a] = __shfl(s, 0, 32) + __shfl(s, 16, 32);
  }

  if (lane == 0) {
    float ce = tot[0] / fmaxf(tot[1], 1.0f);
    float u_l = logf(1.0f + tot[2] / (float)BB);
    out[0] = ce + u_l;                 // UL_ALPHA = 1.0
  }
}

// =====================================================================
extern "C" void kernel_launch(void* const* d_in, const int* in_sizes, int n_in,
                              void* d_out, int out_size, void* d_ws, size_t ws_size,
                              hipStream_t stream) {
  (void)in_sizes; (void)n_in; (void)out_size; (void)ws_size;
  const float* logits = (const float*)d_in[0];
  const int* labels = (const int*)d_in[1];
  float* ws = (float*)d_ws;            // 3 * 2048 floats, fully overwritten
  float* out = (float*)d_out;

  ul_row_kernel<<<ROWS, THREADS, 0, stream>>>(logits, labels, ws);
  ul_finalize_kernel<<<1, 32, 0, stream>>>(ws, out);
}